// HelmholtzTrunk_56547539419416
// MI455X (gfx1250) — compile-verified
//
#include <hip/hip_runtime.h>

typedef __attribute__((ext_vector_type(2))) float v2f;
typedef __attribute__((ext_vector_type(4))) float v4f;
typedef __attribute__((ext_vector_type(8))) float v8f;

#define STRIDE 128
#define TS (16 * STRIDE)

static __device__ __forceinline__ v8f wmma_f32_16x16x4(v2f a, v2f b, v8f c) {
  // (neg_a, A, neg_b, B, c_mod, C, reuse_a, reuse_b)
  return __builtin_amdgcn_wmma_f32_16x16x4_f32(false, a, false, b, (short)0, c,
                                               false, false);
}

// One SIREN layer for forward h and three tangent streams.
// out = act(in @ W + b); tangents: cos(u)*(t @ W) [pre: *W0; block: +residual]
static __device__ void layer_pass(const float* __restrict__ W,
                                  const float* __restrict__ bias,
                                  const float* inH, const float* inT0,
                                  const float* inT1, const float* inT2,
                                  float* outH, float* outT0, float* outT1,
                                  float* outT2, int lane, int isPre) {
  const int lm = lane & 15;   // A row / B col / C col within tile
  const int lh = lane >> 4;   // K-half select
  for (int ct = 0; ct < 8; ++ct) {
    const int n0 = ct * 16;
    v8f aH = {0.f, 0.f, 0.f, 0.f, 0.f, 0.f, 0.f, 0.f};
    v8f a0 = aH, a1 = aH, a2 = aH;
    for (int k0 = 0; k0 < 128; k0 += 4) {
      const int kr = k0 + 2 * lh;
      v2f bf;
      bf.x = W[(kr + 0) * 128 + n0 + lm];
      bf.y = W[(kr + 1) * 128 + n0 + lm];
      const v2f fH = *(const v2f*)(inH + lm * STRIDE + kr);
      const v2f f0 = *(const v2f*)(inT0 + lm * STRIDE + kr);
      const v2f f1 = *(const v2f*)(inT1 + lm * STRIDE + kr);
      const v2f f2 = *(const v2f*)(inT2 + lm * STRIDE + kr);
      aH = wmma_f32_16x16x4(fH, bf, aH);  // forward pre-activation
      a0 = wmma_f32_16x16x4(f0, bf, a0);  // tangent d/dx0
      a1 = wmma_f32_16x16x4(f1, bf, a1);  // tangent d/dx1
      a2 = wmma_f32_16x16x4(f2, bf, a2);  // tangent d/dx2
    }
    const float bn = bias[n0 + lm];
#pragma unroll
    for (int r = 0; r < 8; ++r) {
      const int m = r + 8 * lh;
      const int off = m * STRIDE + n0 + lm;
      float u = aH[r] + bn;
      if (isPre) {
        u *= 30.0f;  // SIREN w0
        outH[off] = __sinf(u);
        const float cw = 30.0f * __cosf(u);
        outT0[off] = cw * a0[r];
        outT1[off] = cw * a1[r];
        outT2[off] = cw * a2[r];
      } else {
        const float cu = __cosf(u);
        outH[off] = inH[off] + __sinf(u);
        outT0[off] = inT0[off] + cu * a0[r];
        outT1[off] = inT1[off] + cu * a1[r];
        outT2[off] = inT2[off] + cu * a2[r];
      }
    }
  }
  __syncthreads();
}

__global__ void __launch_bounds__(32) helmholtz_kernel(
    const float* __restrict__ x, const float* __restrict__ w_dir,
    const float* __restrict__ b_rff, const float* __restrict__ log_sigma,
    const float* __restrict__ A_preW, const float* __restrict__ A_preb,
    const float* __restrict__ A_blkW, const float* __restrict__ A_blkb,
    const float* __restrict__ A_outW, const float* __restrict__ P_preW,
    const float* __restrict__ P_preb, const float* __restrict__ P_blkW,
    const float* __restrict__ P_blkb, const float* __restrict__ P_outW,
    float* __restrict__ out) {
  extern __shared__ float smem[];
  float* H0 = smem;            // buffer set 0
  float* T00 = smem + 1 * TS;
  float* T01 = smem + 2 * TS;
  float* T02 = smem + 3 * TS;
  float* H1 = smem + 4 * TS;   // buffer set 1
  float* T10 = smem + 5 * TS;
  float* T11 = smem + 6 * TS;
  float* T12 = smem + 7 * TS;
  float* xs = H1;    // overlap: set1 unused during RFF stage
  float* Wred = H0;  // overlap: set0 unused during A-reduction

  const int lane = threadIdx.x;
  const int p0 = blockIdx.x * 16;  // first point of this tile
  const float sig = __expf(log_sigma[0]);
  const float coef = 0.125f;  // sqrt(2/D), D=128

  float* u_out = out;                // (B,N,3) = 6144 floats
  float* g_out = out + 4 * 512 * 3;  // grad_phi block-diagonal

  for (int trunk = 0; trunk < 2; ++trunk) {
    const float* preW = trunk ? P_preW : A_preW;
    const float* preb = trunk ? P_preb : A_preb;
    const float* blkW = trunk ? P_blkW : A_blkW;
    const float* blkb = trunk ? P_blkb : A_blkb;

    // ---- RFF features + tangent seeds into buffer set 0 ----
    if (lane < 16) {
      xs[lane * 3 + 0] = x[(p0 + lane) * 3 + 0];
      xs[lane * 3 + 1] = x[(p0 + lane) * 3 + 1];
      xs[lane * 3 + 2] = x[(p0 + lane) * 3 + 2];
    }
    __syncthreads();
    for (int idx = lane; idx < 16 * 128; idx += 32) {
      const int p = idx >> 7, c = idx & 127;
      const float w0 = w_dir[c], w1 = w_dir[128 + c], w2 = w_dir[256 + c];
      const float z =
          sig * (xs[p * 3] * w0 + xs[p * 3 + 1] * w1 + xs[p * 3 + 2] * w2) +
          b_rff[c];
      H0[p * STRIDE + c] = coef * __sinf(z);
      const float t = coef * __cosf(z) * sig;  // d(rff)/dx_i = t * w_dir[i,c]
      T00[p * STRIDE + c] = t * w0;
      T01[p * STRIDE + c] = t * w1;
      T02[p * STRIDE + c] = t * w2;
    }
    __syncthreads();

    layer_pass(preW, preb, H0, T00, T01, T02, H1, T10, T11, T12, lane, 1);
    layer_pass(blkW, blkb, H1, T10, T11, T12, H0, T00, T01, T02, lane, 0);
    layer_pass(blkW + 128 * 128, blkb + 128, H0, T00, T01, T02, H1, T10, T11,
               T12, lane, 0);
    // final activations + tangents now in buffer set 1

    if (trunk == 0) {
      // Wred[c][k] = sum_j A_outW[c*24 + 3j + k]  (head forward/bias cancel)
      for (int c = lane * 4; c < lane * 4 + 4; ++c) {
#pragma unroll
        for (int k = 0; k < 3; ++k) {
          float s = 0.f;
#pragma unroll
          for (int j = 0; j < 8; ++j) s += A_outW[c * 24 + 3 * j + k];
          Wred[c * 3 + k] = s;
        }
      }
      __syncthreads();
      if (lane < 16) {
        float G00 = 0, G01 = 0, G02 = 0, G10 = 0, G11 = 0, G12 = 0;
        float G20 = 0, G21 = 0, G22 = 0;
        const int base = lane * STRIDE;
        for (int c = 0; c < 128; ++c) {
          const float t0 = T10[base + c], t1 = T11[base + c],
                      t2 = T12[base + c];
          const float w0 = Wred[c * 3 + 0], w1 = Wred[c * 3 + 1],
                      w2 = Wred[c * 3 + 2];
          G00 += t0 * w0; G01 += t0 * w1; G02 += t0 * w2;
          G10 += t1 * w0; G11 += t1 * w1; G12 += t1 * w2;
          G20 += t2 * w0; G21 += t2 * w1; G22 += t2 * w2;
        }
        (void)G00; (void)G11; (void)G22;
        const int p = p0 + lane;
        u_out[p * 3 + 0] = G12 - G21;  // gAz_y - gAy_z
        u_out[p * 3 + 1] = G20 - G02;  // gAx_z - gAz_x
        u_out[p * 3 + 2] = G01 - G10;  // gAy_x - gAx_y
      }
      __syncthreads();
    } else {
      if (lane < 16) {
        float J[4][3] = {};
        const int base = lane * STRIDE;
        for (int c = 0; c < 128; ++c) {
          const float t0 = T10[base + c], t1 = T11[base + c],
                      t2 = T12[base + c];
#pragma unroll
          for (int s = 0; s < 4; ++s) {
            const float w = P_outW[c * 4 + s];
            J[s][0] += t0 * w;
            J[s][1] += t1 * w;
            J[s][2] += t2 * w;
          }
        }
        const long long p = p0 + lane;
#pragma unroll
        for (int s = 0; s < 4; ++s) {
          const long long row = (p * 4 + s) * 2048 + p;  // diagonal block
          g_out[row * 3 + 0] = J[s][0];
          g_out[row * 3 + 1] = J[s][1];
          g_out[row * 3 + 2] = J[s][2];
        }
      }
    }
  }
}

// 201 MB streaming zero-fill: non-temporal 128-bit stores so the one-shot
// zeros do not cycle through (and dirty) the 192 MB L2 — pure HBM path.
__global__ void zero_kernel(float* __restrict__ out, long long n) {
  const long long n4 = n >> 2;
  v4f* out4 = (v4f*)out;
  long long i = (long long)blockIdx.x * blockDim.x + threadIdx.x;
  const long long stride = (long long)gridDim.x * blockDim.x;
  const v4f z = {0.f, 0.f, 0.f, 0.f};
  for (; i < n4; i += stride) __builtin_nontemporal_store(z, out4 + i);
  if (blockIdx.x == 0 && threadIdx.x == 0)
    for (long long r = n4 << 2; r < n; ++r)
      __builtin_nontemporal_store(0.f, out + r);
}

extern "C" void kernel_launch(void* const* d_in, const int* in_sizes, int n_in,
                              void* d_out, int out_size, void* d_ws,
                              size_t ws_size, hipStream_t stream) {
  (void)in_sizes; (void)n_in; (void)d_ws; (void)ws_size;
  const float* x        = (const float*)d_in[0];
  const float* w_dir    = (const float*)d_in[1];
  const float* b_rff    = (const float*)d_in[2];
  const float* log_sig  = (const float*)d_in[3];
  const float* A_preW   = (const float*)d_in[4];
  const float* A_preb   = (const float*)d_in[5];
  const float* A_blkW   = (const float*)d_in[6];
  const float* A_blkb   = (const float*)d_in[7];
  const float* A_outW   = (const float*)d_in[8];
  const float* P_preW   = (const float*)d_in[10];
  const float* P_preb   = (const float*)d_in[11];
  const float* P_blkW   = (const float*)d_in[12];
  const float* P_blkb   = (const float*)d_in[13];
  const float* P_outW   = (const float*)d_in[14];
  float* out = (float*)d_out;

  // 1) zero the 201 MB output (HBM-bandwidth bound; the real cost here)
  zero_kernel<<<2048, 256, 0, stream>>>(out, (long long)out_size);

  // 2) fused trunks + JVPs: 128 tiles of 16 points, 1 wave each, 64 KB LDS
  helmholtz_kernel<<<128, 32, 8 * TS * sizeof(float), stream>>>(
      x, w_dir, b_rff, log_sig, A_preW, A_preb, A_blkW, A_blkb, A_outW, P_preW,
      P_preb, P_blkW, P_blkb, P_outW, out);
}